// TrainableEMLTreeV2_37443524886677
// MI455X (gfx1250) — compile-verified
//
#include <hip/hip_runtime.h>

// Soft decision tree (DEPTH=6, BATCH=1M). Transcendental-bound. ALL affine-
// in-x terms go through the matrix unit: leaf init (4 WMMAs) plus every
// node's tL = wl0+wl1*x and tR = wr0+wr1*x (8 WMMAs with pre-permuted A
// matrices so lane-half h receives its nodes' t-values in slot order).
// Combines are then fma(c2,child,t) -> 7 VALU each, and only wl2/wr2 remain
// as hoisted per-lane coefficients (no scratch spills under the 256-VGPR cap).
// Levels 5-3 in-lane; 2-1 split across halves; level 0 after SWAPX16.

#define N_LEAVES 64
#define N_INTERNAL 63
#define WAVES_PER_BLOCK 8
#define THREADS (WAVES_PER_BLOCK * 32)

typedef __attribute__((ext_vector_type(2))) float v2f;
typedef __attribute__((ext_vector_type(8))) float v8f;

__device__ __forceinline__ float swap16(float v) {
    int i = __builtin_bit_cast(int, v);
    i = __builtin_amdgcn_ds_swizzle(i, 0x401F); // group-of-32, xor=0x10: SWAPX16
    return __builtin_bit_cast(float, i);
}

__device__ __forceinline__ float fast_exp(float x) {
#if __has_builtin(__builtin_amdgcn_exp2f)
    return __builtin_amdgcn_exp2f(x * 1.44269504088896340736f);  // v_mul + v_exp_f32
#else
    return __expf(x);
#endif
}

__device__ __forceinline__ float fast_log(float x) {
#if __has_builtin(__builtin_amdgcn_logf)
    return __builtin_amdgcn_logf(x) * 0.69314718055994530942f;   // v_log_f32 + v_mul
#else
    return __logf(x);
#endif
}

// t-values come from WMMA; only the child coefficients remain.
__device__ __forceinline__ float combine2(float c2l, float tL, float c2r,
                                          float tR, float left, float right) {
    float tl = __builtin_fmaf(c2l, left,  tL);
    float tr = __builtin_fmaf(c2r, right, tR);
    return fast_exp(tl) - fast_log(fmaxf(tr, 1e-10f));
}

__global__ __launch_bounds__(THREADS, 4) void eml_tree_kernel(
    const float* __restrict__ x, const float* __restrict__ leaf_logits,
    const float* __restrict__ nlL, const float* __restrict__ nlR,
    float* __restrict__ out, int n)
{
    __shared__ float lds_ab[2 * N_LEAVES];      // per leaf: (a1, a0)
    __shared__ float lds_node[8 * N_INTERNAL];  // per node: wl0 wl1 wl2 0 wr0 wr1 wr2 0
    __shared__ float lds_A[8 * 16 * 2];         // 8 permuted A matrices: row m = (c1, c0)

    const int tid = threadIdx.x;
    // ---- softmax precompute ----
    if (tid < N_LEAVES) {
        float l0 = leaf_logits[2 * tid], l1 = leaf_logits[2 * tid + 1];
        float m = fmaxf(l0, l1);
        float e0 = __expf(l0 - m), e1 = __expf(l1 - m);
        float inv = 1.0f / (e0 + e1);
        lds_ab[2 * tid + 0] = e1 * inv;  // a1 (slope)
        lds_ab[2 * tid + 1] = e0 * inv;  // a0 (bias)
    } else if (tid < N_LEAVES + N_INTERNAL) {
        int k = tid - N_LEAVES;
        {
            float a = nlL[3 * k], b = nlL[3 * k + 1], c = nlL[3 * k + 2];
            float m = fmaxf(a, fmaxf(b, c));
            float e0 = __expf(a - m), e1 = __expf(b - m), e2 = __expf(c - m);
            float inv = 1.0f / (e0 + e1 + e2);
            lds_node[8 * k + 0] = e0 * inv;
            lds_node[8 * k + 1] = e1 * inv;
            lds_node[8 * k + 2] = e2 * inv;
            lds_node[8 * k + 3] = 0.0f;
        }
        {
            float a = nlR[3 * k], b = nlR[3 * k + 1], c = nlR[3 * k + 2];
            float m = fmaxf(a, fmaxf(b, c));
            float e0 = __expf(a - m), e1 = __expf(b - m), e2 = __expf(c - m);
            float inv = 1.0f / (e0 + e1 + e2);
            lds_node[8 * k + 4] = e0 * inv;
            lds_node[8 * k + 5] = e1 * inv;
            lds_node[8 * k + 6] = e2 * inv;
            lds_node[8 * k + 7] = 0.0f;
        }
    }
    __syncthreads();

    // ---- build permuted A matrices for the t-value WMMAs ----
    // Row m (h'=m>>3, r=m&7) of matrix w holds (c1, c0) of node nd(w,m):
    //  w=0/2: lvl5 slots 0-7  (wl / wr):  nd = 31 + 8*(r>>2)     + 4h' + (r&3)
    //  w=1/3: lvl5 slots 8-15 (wl / wr):  nd = 31 + 8*(2+(r>>2)) + 4h' + (r&3)
    //  w=4/5: lvl4 slots 0-7  (wl / wr):  nd = 15 + 4*(r>>1) + 2h' + (r&1)
    //  w=6/7: top levels      (wl / wr):  r<4: 7+2r+h'; r=4,5: 3+2h'+(r-4);
    //                                     r=6: 1+h';    r=7: 0
    if (tid < 128) {
        int w = tid >> 4, m = tid & 15;
        int hp = m >> 3, r = m & 7;
        int nd;
        if (w < 4) {
            int gb = (w & 1) * 2;
            nd = 31 + 8 * (gb + (r >> 2)) + 4 * hp + (r & 3);
        } else if (w < 6) {
            nd = 15 + 4 * (r >> 1) + 2 * hp + (r & 1);
        } else {
            if (r < 4)      nd = 7 + 2 * r + hp;
            else if (r < 6) nd = 3 + 2 * hp + (r - 4);
            else if (r == 6) nd = 1 + hp;
            else            nd = 0;
        }
        int off = (w == 2 || w == 3 || w == 5 || w == 7) ? 4 : 0;  // wr vs wl
        lds_A[(w * 16 + m) * 2 + 0] = lds_node[8 * nd + off + 1];  // c1 (x coeff)
        lds_A[(w * 16 + m) * 2 + 1] = lds_node[8 * nd + off + 0];  // c0 (bias)
    }
    __syncthreads();

    const int lane = tid & 31;
    const int wave = tid >> 5;
    const int h = lane >> 4;
    const int l = lane & 15;
    const int tiles = (n + 15) >> 4;

    // Loop-invariant A matrices (lanes >=16 feed K=2,3 * zero B rows: unmasked).
    v2f Ag[4];
    #pragma unroll
    for (int g = 0; g < 4; ++g) {
        Ag[g].x = lds_ab[2 * (16 * g + l) + 0];
        Ag[g].y = lds_ab[2 * (16 * g + l) + 1];
    }
    v2f AW[8];
    #pragma unroll
    for (int w = 0; w < 8; ++w) {
        AW[w].x = lds_A[(w * 16 + l) * 2 + 0];
        AW[w].y = lds_A[(w * 16 + l) * 2 + 1];
    }
    const float By = h ? 0.0f : 1.0f;

    for (int tile = blockIdx.x * WAVES_PER_BLOCK + wave; tile < tiles;
         tile += gridDim.x * WAVES_PER_BLOCK) {
        const int base = tile << 4;
        const int bi = base + l;
        const int bic = bi < n ? bi : n - 1;   // clamped unconditional load
        const float xb = x[bic];

        v2f B;                  // col n = [x_n, 1, 0, 0]^T
        B.x = h ? 0.0f : xb;
        B.y = By;
        v8f cz = {};

        // Level-5 t-values (tL, tR for the 16 lvl5 nodes per lane, slot order)
        v8f t5L0 = __builtin_amdgcn_wmma_f32_16x16x4_f32(false, AW[0], false, B, (short)0, cz, false, false);
        v8f t5L1 = __builtin_amdgcn_wmma_f32_16x16x4_f32(false, AW[1], false, B, (short)0, cz, false, false);
        v8f t5R0 = __builtin_amdgcn_wmma_f32_16x16x4_f32(false, AW[2], false, B, (short)0, cz, false, false);
        v8f t5R1 = __builtin_amdgcn_wmma_f32_16x16x4_f32(false, AW[3], false, B, (short)0, cz, false, false);

        // Level 5 fused with leaf GEMM: nodes 31 + 8g+4h+p, slot s = 4g+p
        float n5[16];
        #pragma unroll
        for (int g = 0; g < 4; ++g) {
            v8f d = __builtin_amdgcn_wmma_f32_16x16x4_f32(
                false, Ag[g], false, B, (short)0, cz, false, false);
            #pragma unroll
            for (int p = 0; p < 4; ++p) {
                int s = 4 * g + p;
                int nd = 31 + 8 * g + 4 * h + p;
                float tL = (g < 2) ? t5L0[s] : t5L1[s - 8];
                float tR = (g < 2) ? t5R0[s] : t5R1[s - 8];
                n5[s] = combine2(lds_node[8 * nd + 2], tL,
                                 lds_node[8 * nd + 6], tR,
                                 d[2 * p], d[2 * p + 1]);
            }
        }

        // Level 4: nodes 15 + 4g+2h+q, slot s4 = 2g+q
        v8f t4L = __builtin_amdgcn_wmma_f32_16x16x4_f32(false, AW[4], false, B, (short)0, cz, false, false);
        v8f t4R = __builtin_amdgcn_wmma_f32_16x16x4_f32(false, AW[5], false, B, (short)0, cz, false, false);
        float n4[8];
        #pragma unroll
        for (int g = 0; g < 4; ++g)
            #pragma unroll
            for (int q = 0; q < 2; ++q) {
                int s4 = 2 * g + q;
                int nd = 15 + 4 * g + 2 * h + q;
                n4[s4] = combine2(lds_node[8 * nd + 2], t4L[s4],
                                  lds_node[8 * nd + 6], t4R[s4],
                                  n5[4 * g + 2 * q], n5[4 * g + 2 * q + 1]);
            }

        // Top levels t-values: slots 0-3 = lvl3, 4-5 = lvl2, 6 = lvl1, 7 = lvl0
        v8f tTL = __builtin_amdgcn_wmma_f32_16x16x4_f32(false, AW[6], false, B, (short)0, cz, false, false);
        v8f tTR = __builtin_amdgcn_wmma_f32_16x16x4_f32(false, AW[7], false, B, (short)0, cz, false, false);

        // Level 3: nodes 7 + 2g+h; lane half h ends with lvl3 nodes {2g+h}
        float n3[4];
        #pragma unroll
        for (int g = 0; g < 4; ++g) {
            int nd = 7 + 2 * g + h;
            n3[g] = combine2(lds_node[8 * nd + 2], tTL[g],
                             lds_node[8 * nd + 6], tTR[g],
                             n4[2 * g], n4[2 * g + 1]);
        }

        // Level 2 (split): half h computes nodes 2h+q (global 3+2h+q)
        float n2l[2];
        #pragma unroll
        for (int q = 0; q < 2; ++q) {
            float send  = h ? n3[q] : n3[2 + q];
            float recv  = swap16(send);
            float own2  = h ? n3[2 + q] : n3[q];
            float left  = h ? recv : own2;
            float right = h ? own2 : recv;
            int nd = 3 + 2 * h + q;
            n2l[q] = combine2(lds_node[8 * nd + 2], tTL[4 + q],
                              lds_node[8 * nd + 6], tTR[4 + q], left, right);
        }

        // Level 1 (split): node h (global 1+h) from own lvl2 pair
        int nd1 = 1 + h;
        float n1l = combine2(lds_node[8 * nd1 + 2], tTL[6],
                             lds_node[8 * nd1 + 6], tTR[6], n2l[0], n2l[1]);

        // Level 0: exchange; both halves compute root redundantly, h==0 stores
        float r0 = swap16(n1l);
        float left  = h ? r0 : n1l;
        float right = h ? n1l : r0;
        float r = combine2(lds_node[2], tTL[7], lds_node[6], tTR[7], left, right);

        if (h == 0 && bi < n) out[bi] = r;
    }
}

extern "C" void kernel_launch(void* const* d_in, const int* in_sizes, int n_in,
                              void* d_out, int out_size, void* d_ws, size_t ws_size,
                              hipStream_t stream) {
    const float* x    = (const float*)d_in[0];
    const float* leaf = (const float*)d_in[1];
    const float* nlL  = (const float*)d_in[2];
    const float* nlR  = (const float*)d_in[3];
    float* out = (float*)d_out;
    const int n = in_sizes[0];
    if (n <= 0) return;

    int tiles = (n + 15) >> 4;
    int blocks = (tiles + WAVES_PER_BLOCK - 1) / WAVES_PER_BLOCK;
    if (blocks > 2048) blocks = 2048;   // grid-stride; ~4 tiles/wave at 1M batch
    if (blocks < 1) blocks = 1;

    eml_tree_kernel<<<blocks, THREADS, 0, stream>>>(x, leaf, nlL, nlR, out, n);
}